// HeadBase_59639915872654
// MI455X (gfx1250) — compile-verified
//
#include <hip/hip_runtime.h>
#include <hip/hip_bf16.h>
#include <math.h>

// ---------------------------------------------------------------------------
// NTM addressing head for MI455X (gfx1250, wave32).
// Phase 1: fused dot(k, memory-row) + row L2-norm -> cosine sim  [HBM bound]
//          V_WMMA_F32_16X16X4_F32, A fragments loaded directly in WMMA layout
//          (no cross-lane shuffles in the hot loop).
// Phase 2: per-batch softmax + gate + circular 3-tap shift + sharpen + norm.
// ---------------------------------------------------------------------------

typedef float v2f __attribute__((ext_vector_type(2)));
typedef float v8f __attribute__((ext_vector_type(8)));

// ---------------------------------------------------------------------------
// Kernel 1: one wave per 16-row tile of one batch.
//   memory: [B, N, 64] f32, k: [B, 64] f32, sim out: [B, N] f32
// A[16x4] f32 layout (ISA 7.12.2): lanes 0-15: VGPR0=K0,VGPR1=K1 ;
// lanes 16-31: VGPR0=K2,VGPR1=K3 (same row M = lane&15).
// => lane loads float2 at rowbase + 4*kk + (hi?2:0) for each step kk.
// B operand = k broadcast into all 16 columns, so D columns replicate dot[row].
// ---------------------------------------------------------------------------
__global__ __launch_bounds__(256) void ntm_sim_kernel(
    const float* __restrict__ memory,
    const float* __restrict__ k,
    float* __restrict__ sim,
    int N, int tilesPerBatch, int totalWaves) {
  const int lane = threadIdx.x & 31;
  const int wave = blockIdx.x * (blockDim.x >> 5) + (threadIdx.x >> 5);
  if (wave >= totalWaves) return;  // wave-uniform: EXEC stays all-ones below

  const int b    = wave / tilesPerBatch;
  const int tile = wave - b * tilesPerBatch;
  const int n0   = tile << 4;
  const int row  = lane & 15;
  const bool hi  = lane >= 16;
  const int sub  = hi ? 2 : 0;   // K sub-offset within each 4-wide chunk

  // --- load A fragments directly in WMMA layout: 16 x global_load_b64 ---
  const float* mrow = memory + ((size_t)b * N + (n0 + row)) * 64;
  float2 a[16];
#pragma unroll
  for (int kk = 0; kk < 16; ++kk)
    a[kk] = *(const float2*)(mrow + kk * 4 + sub);

  // --- row sum of squares: this lane holds K = {sub, sub+1} mod 4 of all
  //     chunks; partner half (xor 16) holds the other two -> butterfly add.
  float msq = 0.f;
#pragma unroll
  for (int kk = 0; kk < 16; ++kk)
    msq += a[kk].x * a[kk].x + a[kk].y * a[kk].y;
  msq += __shfl_xor(msq, 16, 32);  // all lanes: sumsq of row (lane&15)

  // --- k fragments (B operand values), same split ---
  const float* kb = k + (size_t)b * 64;
  float2 kf[16];
#pragma unroll
  for (int kk = 0; kk < 16; ++kk)
    kf[kk] = *(const float2*)(kb + kk * 4 + sub);

  float ksq = 0.f;
#pragma unroll
  for (int kk = 0; kk < 16; ++kk)
    ksq += kf[kk].x * kf[kk].x + kf[kk].y * kf[kk].y;
  ksq += __shfl_xor(ksq, 16, 32);
  const float k_norm = fmaxf(sqrtf(ksq), 1e-8f);

#if __has_builtin(__builtin_amdgcn_wmma_f32_16x16x4_f32)
  v8f acc = {0.f, 0.f, 0.f, 0.f, 0.f, 0.f, 0.f, 0.f};
#pragma unroll
  for (int kk = 0; kk < 16; ++kk) {
    v2f A  = {a[kk].x, a[kk].y};
    v2f Bv = {kf[kk].x, kf[kk].y};  // k broadcast: every column identical
    acc = __builtin_amdgcn_wmma_f32_16x16x4_f32(
        false, A, false, Bv, (short)0, acc, false, false);
  }
  // D layout: VGPR r -> M = r (lanes 0-15) / M = 8+r (lanes 16-31), value
  // replicated over columns. Writers: lanes 0-7 emit rows n0+lane,
  // lanes 16-23 emit rows n0+8+(lane-16).
  const int r = lane & 7;
  float dotv = acc[0];
#pragma unroll
  for (int i = 1; i < 8; ++i) dotv = (r == i) ? acc[i] : dotv;

  const float msqsel = __shfl(msq, hi ? (lane - 8) : lane, 32);
  if ((lane & 8) == 0) {
    const int rw = (hi ? 8 : 0) + r;
    const float m_norm = fmaxf(sqrtf(msqsel), 1e-8f);
    sim[(size_t)b * N + n0 + rw] = dotv / (k_norm * m_norm);
  }
#else
  // VALU fallback: per-lane partial dot over its K-subset, butterfly combine.
  float dpart = 0.f;
#pragma unroll
  for (int kk = 0; kk < 16; ++kk)
    dpart += a[kk].x * kf[kk].x + a[kk].y * kf[kk].y;
  dpart += __shfl_xor(dpart, 16, 32);
  if (!hi) {
    const float m_norm = fmaxf(sqrtf(msq), 1e-8f);
    sim[(size_t)b * N + n0 + row] = dpart / (k_norm * m_norm);
  }
#endif
}

// ---------------------------------------------------------------------------
// Block reductions (256 threads = 8 waves, wave32)
// ---------------------------------------------------------------------------
__device__ __forceinline__ float blockSum256(float v, volatile float* red) {
#pragma unroll
  for (int o = 16; o > 0; o >>= 1) v += __shfl_xor(v, o, 32);
  if ((threadIdx.x & 31) == 0) red[threadIdx.x >> 5] = v;
  __syncthreads();
  float r = red[0];
#pragma unroll
  for (int i = 1; i < 8; ++i) r += red[i];
  __syncthreads();
  return r;
}

__device__ __forceinline__ float blockMax256(float v, volatile float* red) {
#pragma unroll
  for (int o = 16; o > 0; o >>= 1) v = fmaxf(v, __shfl_xor(v, o, 32));
  if ((threadIdx.x & 31) == 0) red[threadIdx.x >> 5] = v;
  __syncthreads();
  float r = red[0];
#pragma unroll
  for (int i = 1; i < 8; ++i) r = fmaxf(r, red[i]);
  __syncthreads();
  return r;
}

// ---------------------------------------------------------------------------
// Kernel 2: one block per batch. softmax(beta*sim) -> gate -> circular 3-tap
// shift (kernel s) -> pow(gamma) -> normalize.  N <= 4096 staged in LDS.
// ---------------------------------------------------------------------------
__global__ __launch_bounds__(256) void ntm_addr_kernel(
    const float* __restrict__ sim,
    const float* __restrict__ beta_p,
    const float* __restrict__ prev_w,
    const float* __restrict__ g_p,
    const float* __restrict__ s_p,
    const float* __restrict__ gamma_p,
    float* __restrict__ out,
    int N) {
  __shared__ float wgl[4096];
  __shared__ float red[8];

  const int b   = blockIdx.x;
  const int tid = threadIdx.x;
  const int npt = N >> 8;  // elements per thread (16 for N=4096)

  const float beta  = beta_p[b];
  const float g     = g_p[b];
  const float gamma = gamma_p[b];
  const float s0 = s_p[b * 3 + 0];
  const float s1 = s_p[b * 3 + 1];
  const float s2 = s_p[b * 3 + 2];

  const float* simb = sim + (size_t)b * N;
  const float* pwb  = prev_w + (size_t)b * N;

  // 1) z = beta*sim, block max
  float zv[16];
  float lmax = -INFINITY;
#pragma unroll 4
  for (int j = 0; j < npt; ++j) {
    zv[j] = beta * simb[tid + (j << 8)];
    lmax = fmaxf(lmax, zv[j]);
  }
  const float mx = blockMax256(lmax, red);

  // 2) exp + sum
  float lsum = 0.f;
#pragma unroll 4
  for (int j = 0; j < npt; ++j) {
    zv[j] = __expf(zv[j] - mx);
    lsum += zv[j];
  }
  const float inv = 1.f / blockSum256(lsum, red);

  // 3) gate with prev_w, stage w_g in LDS
#pragma unroll 4
  for (int j = 0; j < npt; ++j) {
    const int i = tid + (j << 8);
    const float wc = zv[j] * inv;
    wgl[i] = g * wc + (1.f - g) * pwb[i];
  }
  __syncthreads();

  // 4) circular 3-tap shift + sharpen, sum
  float wp[16];
  float lsum2 = 0.f;
#pragma unroll 4
  for (int j = 0; j < npt; ++j) {
    const int i  = tid + (j << 8);
    const int im = (i == 0) ? (N - 1) : (i - 1);
    const int ip = (i == N - 1) ? 0 : (i + 1);
    const float ws = s0 * wgl[im] + s1 * wgl[i] + s2 * wgl[ip];
    const float p  = powf(ws, gamma);
    wp[j] = p;
    lsum2 += p;
  }
  const float invt = 1.f / (blockSum256(lsum2, red) + 1e-16f);

  // 5) normalize
#pragma unroll 4
  for (int j = 0; j < npt; ++j) out[(size_t)b * N + tid + (j << 8)] = wp[j] * invt;
}

// ---------------------------------------------------------------------------
extern "C" void kernel_launch(void* const* d_in, const int* in_sizes, int n_in,
                              void* d_out, int out_size, void* d_ws,
                              size_t ws_size, hipStream_t stream) {
  (void)n_in; (void)out_size; (void)ws_size;
  const float* memory = (const float*)d_in[0];
  const float* k      = (const float*)d_in[1];
  const float* beta   = (const float*)d_in[2];
  const float* prev_w = (const float*)d_in[3];
  const float* g      = (const float*)d_in[4];
  const float* s      = (const float*)d_in[5];
  const float* gamma  = (const float*)d_in[6];

  const int B = in_sizes[2];            // beta is [B,1]
  const int N = in_sizes[3] / B;        // prev_w is [B,N]   (4096)
  // in_sizes[1]/B == M == 64 (layout hardcoded in kernel 1)

  float* sim = (float*)d_ws;            // [B,N] f32 scratch (4 MB)

  const int tilesPerBatch = N >> 4;
  const int totalWaves = B * tilesPerBatch;          // 65536
  const int blocks = (totalWaves + 7) >> 3;          // 8 waves / block

  ntm_sim_kernel<<<blocks, 256, 0, stream>>>(memory, k, sim, N, tilesPerBatch,
                                             totalWaves);
  ntm_addr_kernel<<<B, 256, 0, stream>>>(sim, beta, prev_w, g, s, gamma,
                                         (float*)d_out, N);
}